// QNN_57286273794856
// MI455X (gfx1250) — compile-verified
//
#include <hip/hip_runtime.h>
#include <math.h>

typedef __attribute__((ext_vector_type(2))) float v2f;
typedef __attribute__((ext_vector_type(8))) float v8f;

#define N_QUBITS 4
#define DIM 16
#define BATCH 524288

// lane-XOR gather via ds_swizzle_b32 (group-of-32: and=0x1f, or=0, xor=M)
template <int M>
__device__ __forceinline__ float swz_xor(float v) {
    return __int_as_float(
        __builtin_amdgcn_ds_swizzle(__float_as_int(v), 0x1f | (M << 10)));
}

// ---------------------------------------------------------------------------
// Kernel 1: build the fixed 16x16 trainable-circuit matrix U (row-major,
// U[i][j] = <i| T |j>), where T = RY(w2) * ring * RY(w1) * ring * RY(w0).
// Wire 0 is the most-significant bit of the state index (bit value 8>>q).
// ---------------------------------------------------------------------------
__global__ __launch_bounds__(16) void qnn_build_unitary(
        const float* __restrict__ w, float* __restrict__ U) {
    const int j = threadIdx.x;   // basis column, 16 threads
    float st[DIM];
#pragma unroll
    for (int i = 0; i < DIM; ++i) st[i] = (i == j) ? 1.0f : 0.0f;

#pragma unroll
    for (int layer = 0; layer < 3; ++layer) {
#pragma unroll
        for (int q = 0; q < N_QUBITS; ++q) {
            float c, s;
            __sincosf(w[layer * 4 + q] * 0.5f, &s, &c);
            const int bv = 8 >> q;
#pragma unroll
            for (int i = 0; i < DIM; ++i) {
                if (i & bv) continue;
                const float a = st[i], bb = st[i | bv];
                st[i]      = c * a - s * bb;   // RY = [[c,-s],[s,c]]
                st[i | bv] = s * a + c * bb;
            }
        }
        if (layer < 2) {
            // ring CNOTs: (0,1),(1,2),(2,3),(3,0); control=1 flips target
#pragma unroll
            for (int e = 0; e < 4; ++e) {
                const int bc = 8 >> e;
                const int bt = 8 >> ((e + 1) & 3);
#pragma unroll
                for (int i = 0; i < DIM; ++i) {
                    if ((i & bc) && !(i & bt)) {
                        const float tmp = st[i];
                        st[i] = st[i | bt];
                        st[i | bt] = tmp;
                    }
                }
            }
        }
    }
#pragma unroll
    for (int i = 0; i < DIM; ++i) U[i * DIM + j] = st[i];
}

// ---------------------------------------------------------------------------
// Kernel 2: one wave32 handles 16 samples.
//   S_final(16x16) = S_enc(16x16) * U^T  via 4x V_WMMA_F32_16X16X4_F32
// A fragment (MxK): lane L -> sample m=L&15, K = 4t + 2*hi + {0,1}
// B fragment (KxN): lane L -> col n=L&15, rows K = 4t + 2*hi + {0,1}
//                   => contiguous float2 at U[n][4t + 2*hi]
// D layout: acc[r] on lane L = S_final[sample r + 8*hi][state n=L&15]
// ---------------------------------------------------------------------------
__global__ __launch_bounds__(256) void qnn_main(const float* __restrict__ x,
                                                const float* __restrict__ U,
                                                const float* __restrict__ W,
                                                const float* __restrict__ b,
                                                float* __restrict__ out) {
    const int lane = threadIdx.x & 31;
    const int m    = lane & 15;        // sample row within tile / state col
    const int hi   = lane >> 4;        // half-wave select
    const int wave = blockIdx.x * (blockDim.x >> 5) + (threadIdx.x >> 5);

    // ---- per-sample encoding angles (product state after 4 RYs on |0000>)
    const float4 xa = ((const float4*)x)[wave * 16 + m];
    float c0, s0, c1, s1, c2, s2, c3, s3;
    __sincosf(xa.x * 0.5f, &s0, &c0);
    __sincosf(xa.y * 0.5f, &s1, &c1);
    __sincosf(xa.z * 0.5f, &s2, &c2);
    __sincosf(xa.w * 0.5f, &s3, &c3);

    // amp(j = b0 b1 b2 b3) = f0(b0) f1(b1) f2(b2) f3(b3); this lane owns
    // indices with b2 == hi, b0b1 == t, b3 in {0,1}.
    const float t2 = hi ? s2 : c2;
    float pre[4];
    pre[0] = c0 * c1 * t2;
    pre[1] = c0 * s1 * t2;
    pre[2] = s0 * c1 * t2;
    pre[3] = s0 * s1 * t2;

    const float* urow = U + m * DIM + 2 * hi;
    v8f acc = {};
#pragma unroll
    for (int t = 0; t < 4; ++t) {
        v2f A;
        A.x = pre[t] * c3;
        A.y = pre[t] * s3;
        const v2f B = *(const v2f*)(urow + 4 * t);   // 8B-aligned
        acc = __builtin_amdgcn_wmma_f32_16x16x4_f32(
            false, A, false, B, (short)0, acc, false, false);
    }

    // ---- probabilities + <Z_q> via signed 16-lane butterflies.
    // Qubit q corresponds to state-index bit (8>>q) == lane bit (8>>q).
    // Each lane keeps the z-tuple of the sample r == (lane & 7); after the
    // loop, lanes with (lane&8)==0 own one distinct sample each.
    float zq0 = 0.0f, zq1 = 0.0f, zq2 = 0.0f, zq3 = 0.0f;
#pragma unroll
    for (int r = 0; r < 8; ++r) {
        const float p = acc[r] * acc[r];

        const float x8 = swz_xor<8>(p);
        const float sA = p + x8;
        float d0 = p - x8;
        d0 += swz_xor<4>(d0); d0 += swz_xor<2>(d0); d0 += swz_xor<1>(d0);

        const float x4 = swz_xor<4>(sA);
        const float sB = sA + x4;
        float d1 = sA - x4;
        d1 += swz_xor<2>(d1); d1 += swz_xor<1>(d1);

        const float x2 = swz_xor<2>(sB);
        const float sC = sB + x2;
        float d2 = sB - x2;
        d2 += swz_xor<1>(d2);

        const float d3 = sC - swz_xor<1>(sC);

        const bool sel = (lane & 7) == r;
        zq0 = sel ? d0 : zq0;
        zq1 = sel ? d1 : zq1;
        zq2 = sel ? d2 : zq2;
        zq3 = sel ? d3 : zq3;
    }
    // Subtract-butterflies are negated on lanes where the bit is set.
    // Store lanes have (lane&8)==0 so zq0 needs no fixup.
    zq1 = (lane & 4) ? -zq1 : zq1;
    zq2 = (lane & 2) ? -zq2 : zq2;
    zq3 = (lane & 1) ? -zq3 : zq3;

    // ---- linear head + softmax: 16 lanes, one sample each, single pass
    if ((lane & 8) == 0) {
        float lg[6];
#pragma unroll
        for (int k = 0; k < 6; ++k) {
            lg[k] = fmaf(W[k * 4 + 0], zq0,
                    fmaf(W[k * 4 + 1], zq1,
                    fmaf(W[k * 4 + 2], zq2,
                    fmaf(W[k * 4 + 3], zq3, b[k]))));
        }
        float mx = lg[0];
#pragma unroll
        for (int k = 1; k < 6; ++k) mx = fmaxf(mx, lg[k]);
        float e[6], ssum = 0.0f;
#pragma unroll
        for (int k = 0; k < 6; ++k) { e[k] = __expf(lg[k] - mx); ssum += e[k]; }
        const float inv = 1.0f / ssum;

        const size_t samp = (size_t)(wave * 16 + (lane & 7) + 8 * hi);
        float* o = out + samp * 6;               // 24B stride -> 8B aligned
        float2 p01 = {e[0] * inv, e[1] * inv};
        float2 p23 = {e[2] * inv, e[3] * inv};
        float2 p45 = {e[4] * inv, e[5] * inv};
        *(float2*)(o + 0) = p01;
        *(float2*)(o + 2) = p23;
        *(float2*)(o + 4) = p45;
    }
}

extern "C" void kernel_launch(void* const* d_in, const int* in_sizes, int n_in,
                              void* d_out, int out_size, void* d_ws, size_t ws_size,
                              hipStream_t stream) {
    const float* x  = (const float*)d_in[0];   // [BATCH,4]
    const float* qw = (const float*)d_in[1];   // [3,4]
    const float* W  = (const float*)d_in[2];   // [6,4]
    const float* b  = (const float*)d_in[3];   // [6]
    float* out = (float*)d_out;                // [BATCH,6]
    float* U   = (float*)d_ws;                 // 16*16 floats

    qnn_build_unitary<<<1, 16, 0, stream>>>(qw, U);

    const int waves  = BATCH / 16;             // 32768
    const int block  = 256;                    // 8 waves
    const int grid   = waves / (block / 32);   // 4096
    qnn_main<<<grid, block, 0, stream>>>(x, U, W, b, out);
}